// ChannelMerger_73916387164746
// MI455X (gfx1250) — compile-verified
//
#include <hip/hip_runtime.h>

typedef __attribute__((ext_vector_type(2))) float v2f;
typedef __attribute__((ext_vector_type(8))) float v8f;
typedef __attribute__((address_space(3))) float lds_f;

namespace {
constexpr int Bn = 64, Cn = 273, Tn = 2048, On = 256, Dn = 2048;
constexpr int CPAD = 288;                       // C padded to 18*16 for WMMA K-loop
constexpr int XS = 144;                         // x-tile row stride (dwords): 2*144%64==32 -> conflict-free halves; %4==0 -> 16B aligned rows
constexpr float MARGIN = 0.2f;
constexpr float W2PI = 4.48798950512827605495f; // 2*pi / (1 + 2*margin)
}

__device__ __forceinline__ v8f wmma4(v2f a, v2f b, v8f c) {
  // D = A(16x4 f32) * B(4x16 f32) + C(16x16 f32)
  return __builtin_amdgcn_wmma_f32_16x16x4_f32(false, a, false, b, (short)0, c,
                                               false, false);
}

// ---------------------------------------------------------------------------
// Kernel 1: fused fourier embedding + scores GEMM.
// scores[bc, o] = sum_d emb[bc, d] * heads[o, d]
// Block: 256 thr (8 waves), owns 16 bc rows, full O=256 (2 o-tiles per wave).
// Embedding chunk (16 rows x 256 K) generated into LDS, transposed + xor-swizzled.
// ---------------------------------------------------------------------------
__global__ __launch_bounds__(256) void cm_scores(const float* __restrict__ pos,
                                                 const float* __restrict__ heads,
                                                 float* __restrict__ scores) {
  __shared__ float embT[256 * 16]; // [kc][16], addr = kc*16 + (row ^ (kc&15))
  const int bc0 = blockIdx.x * 16;
  const int tid = threadIdx.x;
  const int lane = tid & 31;
  const int wv = tid >> 5;          // 0..7
  const int m = lane & 15;          // A-matrix row / B-matrix col
  const int hk = (lane >> 4) << 1;  // K sub-offset: 0 (lanes 0-15) or 2 (16-31)
  const int o0a = wv * 16;
  const int o0b = (wv + 8) * 16;

  float X[16], Y[16];
#pragma unroll
  for (int i = 0; i < 16; ++i) {
    X[i] = pos[(bc0 + i) * 2 + 0] + MARGIN;
    Y[i] = pos[(bc0 + i) * 2 + 1] + MARGIN;
  }

  v8f accA, accB;
#pragma unroll
  for (int q = 0; q < 8; ++q) { accA[q] = 0.0f; accB[q] = 0.0f; }

  for (int ch = 0; ch < 8; ++ch) {      // 8 chunks of K=256 (d = ch*256 + tid)
    const bool isCos = ch < 4;          // d < 1024 -> cos part, else sin part
    const int idx2 = (ch & 3) * 256 + tid;
    const float fx = (float)(idx2 >> 5) * W2PI;
    const float fy = (float)(idx2 & 31) * W2PI;
    __syncthreads();
#pragma unroll
    for (int i = 0; i < 16; ++i) {
      const float ang = X[i] * fx + Y[i] * fy;
      embT[tid * 16 + (i ^ (tid & 15))] = isCos ? __cosf(ang) : __sinf(ang);
    }
    __syncthreads();
    const int d0 = ch * 256;
#pragma unroll 4
    for (int kk = 0; kk < 64; ++kk) {
      const int kc = kk * 4 + hk;
      v2f a;
      a.x = embT[kc * 16 + (m ^ (kc & 15))];
      a.y = embT[(kc + 1) * 16 + (m ^ ((kc + 1) & 15))];
      const int dk = d0 + kc;
      const v2f bA = *(const v2f*)(heads + (o0a + m) * Dn + dk);
      const v2f bB = *(const v2f*)(heads + (o0b + m) * Dn + dk);
      accA = wmma4(a, bA, accA);
      accB = wmma4(a, bB, accB);
    }
  }

  // C/D layout: VGPR r -> M=r (lanes 0-15), M=r+8 (lanes 16-31); N = lane&15
  const int mb = (lane >> 4) << 3;
#pragma unroll
  for (int r = 0; r < 8; ++r) {
    const int bc = bc0 + mb + r;
    scores[(size_t)bc * On + o0a + m] = accA[r];
    scores[(size_t)bc * On + o0b + m] = accB[r];
  }
}

// ---------------------------------------------------------------------------
// Kernel 2: masked softmax over C per (b,o) row; writes zero-padded weights
// wts layout: (B, O, CPAD=288), padding columns are exactly 0.
// One wave per row, 9 elements per lane.
// ---------------------------------------------------------------------------
__global__ __launch_bounds__(256) void cm_softmax(const float* __restrict__ scores,
                                                  const unsigned char* __restrict__ invalid,
                                                  float* __restrict__ wts) {
  const int row = blockIdx.x * 8 + (threadIdx.x >> 5); // b*256 + o
  const int lane = threadIdx.x & 31;
  const int b = row >> 8;
  const int o = row & 255;

  float v[9];
  float mx = -__builtin_inff();
#pragma unroll
  for (int j = 0; j < 9; ++j) {
    const int c = lane + j * 32;
    float s = -__builtin_inff();
    if (c < Cn) {
      s = scores[(size_t)(b * Cn + c) * On + o];
      if (invalid[b * Cn + c]) s = -__builtin_inff();
    }
    v[j] = s;
    mx = fmaxf(mx, s);
  }
#pragma unroll
  for (int off = 16; off > 0; off >>= 1) mx = fmaxf(mx, __shfl_xor(mx, off, 32));
  float sum = 0.0f;
#pragma unroll
  for (int j = 0; j < 9; ++j) {
    const float e = __expf(v[j] - mx); // exp(-inf - mx) == 0 -> padding weight 0
    v[j] = e;
    sum += e;
  }
#pragma unroll
  for (int off = 16; off > 0; off >>= 1) sum += __shfl_xor(sum, off, 32);
  const float inv = 1.0f / sum;
#pragma unroll
  for (int j = 0; j < 9; ++j)
    wts[(size_t)row * CPAD + lane + j * 32] = v[j] * inv; // c = lane+32j < 288
}

// ---------------------------------------------------------------------------
// Kernel 3: out[b,o,t] = sum_c wts[b,o,c] * x[b,c,t]   (K = 288, zero padded)
// Block: 512 thr = 16 waves = all 256 o rows; t-tile = 128.
// x tiles are DMA'd into LDS with GLOBAL_LOAD_ASYNC_TO_LDS_B128 (ASYNCcnt),
// double-buffered so the next tile copy overlaps the current WMMA loop.
// ---------------------------------------------------------------------------
__global__ __launch_bounds__(512) void cm_merge(const float* __restrict__ x,
                                                const float* __restrict__ wts,
                                                float* __restrict__ out) {
  __shared__ __align__(16) float xt[2][16 * XS];
  const int b = blockIdx.y;
  const int t0 = blockIdx.x * 128;
  const int tid = threadIdx.x;
  const int lane = tid & 31;
  const int wv = tid >> 5;          // 0..15
  const int o0 = wv * 16;
  const int m = lane & 15;
  const int hk = (lane >> 4) << 1;

  v8f acc[8];
#pragma unroll
  for (int t = 0; t < 8; ++t)
#pragma unroll
    for (int q = 0; q < 8; ++q) acc[t][q] = 0.0f;

  const float* xb = x + (size_t)b * Cn * Tn;
  const float* wrow = wts + (size_t)(b * On + o0 + m) * CPAD;

  // Async tile loader: thread tid copies one 16B chunk of the 16x128 tile.
  const int lr = tid >> 5;         // tile row 0..15
  const int lc = (tid & 31) * 4;   // tile col 0..124 (16B chunks)
  auto issue_tile = [&](int p, int ch) {
    int c = ch * 16 + lr;
    c = (c > Cn - 1) ? (Cn - 1) : c;       // clamp; padded weights are zero
    const float* gp = xb + (size_t)c * Tn + t0 + lc;
    lds_f* lp = (lds_f*)&xt[p][lr * XS + lc];
    asm volatile("global_load_async_to_lds_b128 %0, %1, off"
                 :: "v"(lp), "v"(gp) : "memory");
  };

  issue_tile(0, 0);

  for (int ch = 0; ch < 18; ++ch) {   // K chunks of 16 (18*16 == CPAD)
    const int p = ch & 1;
    asm volatile("s_wait_asynccnt 0x0" ::: "memory"); // my tile writes landed
    __syncthreads();                                  // all waves' writes landed;
                                                      // all done reading xt[p^1]
    if (ch + 1 < 18) issue_tile(p ^ 1, ch + 1);
    const int c0 = ch * 16;
    const float* xtp = &xt[p][0];
#pragma unroll
    for (int ks = 0; ks < 4; ++ks) {
      const int kb = ks * 4 + hk;
      const v2f a = *(const v2f*)(wrow + c0 + kb); // W[m, k..k+1], 8B aligned
      const float* xrow = &xtp[kb * XS + m];
#pragma unroll
      for (int tt = 0; tt < 8; ++tt) {
        v2f bv;
        bv.x = xrow[tt * 16];         // x[k,   t0 + tt*16 + m]
        bv.y = xrow[XS + tt * 16];    // x[k+1, ...]
        acc[tt] = wmma4(a, bv, acc[tt]);
      }
    }
  }

  float* ob = out + (size_t)(b * On + o0) * Tn + t0;
  const int mb = (lane >> 4) << 3;
#pragma unroll
  for (int tt = 0; tt < 8; ++tt)
#pragma unroll
    for (int r = 0; r < 8; ++r)
      ob[(size_t)(mb + r) * Tn + tt * 16 + m] = acc[tt][r];
}

// ---------------------------------------------------------------------------
extern "C" void kernel_launch(void* const* d_in, const int* in_sizes, int n_in,
                              void* d_out, int out_size, void* d_ws, size_t ws_size,
                              hipStream_t stream) {
  const float* x = (const float*)d_in[0];              // (B, C, T) f32
  const float* positions = (const float*)d_in[1];      // (B, C, 2) f32
  const unsigned char* invalid = (const unsigned char*)d_in[2]; // (B, C) bool
  const float* heads = (const float*)d_in[3];          // (O, D) f32
  float* out = (float*)d_out;                          // (B, O, T) f32

  float* scores = (float*)d_ws;                        // (B*C, O) f32 : 17.9 MB
  float* wtsbuf = scores + (size_t)Bn * Cn * On;       // (B, O, CPAD) : 18.9 MB

  cm_scores<<<dim3((Bn * Cn) / 16), dim3(256), 0, stream>>>(positions, heads, scores);
  cm_softmax<<<dim3((Bn * On) / 8), dim3(256), 0, stream>>>(scores, invalid, wtsbuf);
  cm_merge<<<dim3(Tn / 128, Bn), dim3(512), 0, stream>>>(x, wtsbuf, out);
}